// QueryAndGroup_87101936763058
// MI455X (gfx1250) — compile-verified
//
#include <hip/hip_runtime.h>

// ---------------- problem constants (from reference setup_inputs) -----------
#define BATCH 4
#define NPTS  16384
#define MQ    4096
#define KNN   32
#define CHUNK 1024         // points per staged chunk (12 KB of float3)
#define NCHUNK (NPTS / CHUNK)
#define WGQ   64           // queries per workgroup (2 waves x 32)

typedef __attribute__((ext_vector_type(2))) float v2f;
typedef __attribute__((ext_vector_type(8))) float v8f;

// ============================================================================
// Kernel 1: exact KNN via V_WMMA_F32_16X16X4_F32 distance tiles.
//   d'(q,p) = |p|^2 - 2 q.p   (|q|^2 dropped: constant per row, order-invariant)
//   A row m = (-2qx, -2qy, -2qz, 1) ; B col n = (px, py, pz, |p|^2) ; K = 4
// Each wave owns 32 queries (one per lane). Point chunks are double-buffered
// into LDS with GLOBAL_LOAD_ASYNC_TO_LDS_B128 (ASYNCcnt) so the next chunk
// streams in while the current chunk's 64 WMMA tiles are processed.
// ============================================================================
__global__ __launch_bounds__(WGQ) void knn_kernel(const float* __restrict__ xyz,
                                                  const float* __restrict__ new_xyz,
                                                  int* __restrict__ idx_out) {
  __shared__ float sbuf[2][CHUNK * 3];  // double-buffered raw xyz (2 x 12 KB)
  __shared__ float xpose[2][32 * 16];   // per-wave transpose buffer (2 KB each)

  const int tid  = threadIdx.x;
  const int lane = tid & 31;
  const int wave = tid >> 5;
  const int nblk = MQ / WGQ;
  const int b     = blockIdx.x / nblk;
  const int qbase = (blockIdx.x % nblk) * WGQ + wave * 32;  // this wave's 32 queries

  // ---- loop-invariant A fragments (two 16-row M-tiles) ----
  // A layout (16x4 f32): lanes 0-15: row M=lane, v0=K0, v1=K1
  //                      lanes 16-31: row M=lane-16, v0=K2, v1=K3
  const float* qp = new_xyz + ((size_t)b * MQ + qbase) * 3;
  const int rsel = lane & 15;
  float a0x, a0y, a1x, a1y;
  if (lane < 16) {
    a0x = -2.0f * qp[rsel * 3 + 0];        // K0 = -2qx
    a0y = -2.0f * qp[rsel * 3 + 1];        // K1 = -2qy
    a1x = -2.0f * qp[(16 + rsel) * 3 + 0];
    a1y = -2.0f * qp[(16 + rsel) * 3 + 1];
  } else {
    a0x = -2.0f * qp[rsel * 3 + 2];        // K2 = -2qz
    a0y = 1.0f;                            // K3 = 1
    a1x = -2.0f * qp[(16 + rsel) * 3 + 2];
    a1y = 1.0f;
  }
  v2f av0; av0.x = a0x; av0.y = a0y;
  v2f av1; av1.x = a1x; av1.y = a1y;

  // ---- per-lane top-32 (unsorted, tracked max) ----
  float dist[KNN];
  int   idxv[KNN];
#pragma unroll
  for (int j = 0; j < KNN; ++j) { dist[j] = __builtin_inff(); idxv[j] = 0; }
  float dmax = __builtin_inff();

  float* xb = xpose[wave];
  const int col    = lane & 15;
  const int rowAdd = (lane >> 4) * 8;       // C layout: rows v (+8 for hi lanes)

  const char* gb_batch = (const char*)(xyz + (size_t)b * NPTS * 3);

  // async prefetch of one chunk into sbuf[s]: 768 B128 transfers, 12 per thread
  auto issue_chunk = [&](int c, int s) {
    const char* gbase = gb_batch + (size_t)c * CHUNK * 12;  // 16B aligned
    unsigned lbase = (unsigned)(uintptr_t)(&sbuf[s][0]);
    for (int i = tid; i < (CHUNK * 3) / 4; i += WGQ) {
      unsigned la  = lbase + (unsigned)i * 16u;
      unsigned long long ga = (unsigned long long)(uintptr_t)(gbase + (size_t)i * 16);
      asm volatile("global_load_async_to_lds_b128 %0, %1, off"
                   :: "v"(la), "v"(ga) : "memory");
    }
  };

  issue_chunk(0, 0);

  for (int c = 0; c < NCHUNK; ++c) {
    asm volatile("s_wait_asynccnt 0" ::: "memory");  // my slice of chunk c done
    __syncthreads();  // all waves' slices visible; everyone past chunk c-1
    if (c + 1 < NCHUNK) issue_chunk(c + 1, (c + 1) & 1);  // overlap with compute

    const float* sb = &sbuf[c & 1][0];
    const int cbase = c * CHUNK;

    for (int t = 0; t < CHUNK; t += 16) {
      // ---- B fragment: lane = column (point), halves split K ----
      const float* sp = sb + (t + col) * 3;
      float px = sp[0], py = sp[1], pz = sp[2];
      float pp = fmaf(px, px, fmaf(py, py, pz * pz));
      v2f bv;
      bv.x = (lane < 16) ? px : pz;   // K0 / K2
      bv.y = (lane < 16) ? py : pp;   // K1 / K3

      v8f c0 = {}; v8f c1 = {};
      c0 = __builtin_amdgcn_wmma_f32_16x16x4_f32(false, av0, false, bv,
                                                 (short)0, c0, false, false);
      c1 = __builtin_amdgcn_wmma_f32_16x16x4_f32(false, av1, false, bv,
                                                 (short)0, c1, false, false);

      // ---- transpose through per-wave LDS: [query row 0..31][point col 0..15]
#pragma unroll
      for (int v = 0; v < 8; ++v) {
        xb[(rowAdd + v) * 16 + col]      = c0[v];
        xb[(16 + rowAdd + v) * 16 + col] = c1[v];
      }
      asm volatile("s_wait_dscnt 0" ::: "memory");  // cross-lane LDS visibility (same wave)

      const float4* rp = reinterpret_cast<const float4*>(&xb[lane * 16]);
      float4 q0 = rp[0], q1 = rp[1], q2 = rp[2], q3 = rp[3];
      float rowv[16] = {q0.x, q0.y, q0.z, q0.w, q1.x, q1.y, q1.z, q1.w,
                        q2.x, q2.y, q2.z, q2.w, q3.x, q3.y, q3.z, q3.w};
      asm volatile("" ::: "memory");  // keep next tile's stores after these loads

      const int gb = cbase + t;
#pragma unroll
      for (int cc = 0; cc < 16; ++cc) {
        float d = rowv[cc];
        if (d < dmax) {               // rare path: replace current max
          const int gidx = gb + cc;
          bool done = false;
#pragma unroll
          for (int j = 0; j < KNN; ++j) {
            bool rep = (!done) && (dist[j] == dmax);
            dist[j] = rep ? d : dist[j];
            idxv[j] = rep ? gidx : idxv[j];
            done = done || rep;
          }
          float mx = dist[0];
#pragma unroll
          for (int j = 1; j < KNN; ++j) mx = fmaxf(mx, dist[j]);
          dmax = mx;
        }
      }
    }
  }

  // ---- sort ascending by distance (match jax.lax.top_k ordering) ----
#pragma unroll
  for (int ksz = 2; ksz <= KNN; ksz <<= 1) {
#pragma unroll
    for (int j = ksz >> 1; j > 0; j >>= 1) {
#pragma unroll
      for (int i = 0; i < KNN; ++i) {
        int l = i ^ j;
        if (l > i) {
          float di = dist[i], dl = dist[l];
          int ii = idxv[i], il = idxv[l];
          bool up = ((i & ksz) == 0);
          bool sw = up ? (di > dl) : (di < dl);
          dist[i] = sw ? dl : di; dist[l] = sw ? di : dl;
          idxv[i] = sw ? il : ii; idxv[l] = sw ? ii : il;
        }
      }
    }
  }

  int* wrow = idx_out + ((size_t)b * MQ + qbase + lane) * KNN;
#pragma unroll
  for (int j = 0; j < KNN; ++j) wrow[j] = idxv[j];
}

// ============================================================================
// Kernel 2: gather + concat.
//   out[0 .. B*67*M*K)  = new_features (3 xyz-diff channels + 64 features)
//   out[B*67*M*K .. )   = grouped_xyz
// One thread per (b, m, kk); writes coalesced along kk; feature gathers hit L2.
// ============================================================================
__global__ __launch_bounds__(256) void gather_kernel(const float* __restrict__ xyz,
                                                     const float* __restrict__ new_xyz,
                                                     const float* __restrict__ feat,
                                                     const int* __restrict__ idx,
                                                     float* __restrict__ out) {
  const int b = blockIdx.y;
  const int e = blockIdx.x * blockDim.x + threadIdx.x;  // < MQ*KNN
  const int m  = e >> 5;
  const int kk = e & 31;

  const int p = idx[((size_t)b * MQ + m) * KNN + kk];
  const float* xr = xyz + ((size_t)b * NPTS + p) * 3;
  const float* qr = new_xyz + ((size_t)b * MQ + m) * 3;
  const float xs0 = xr[0], xs1 = xr[1], xs2 = xr[2];

  // new_features channels 0..2 = grouped_xyz - new_xyz
  out[(((size_t)b * 67 + 0) * MQ + m) * KNN + kk] = xs0 - qr[0];
  out[(((size_t)b * 67 + 1) * MQ + m) * KNN + kk] = xs1 - qr[1];
  out[(((size_t)b * 67 + 2) * MQ + m) * KNN + kk] = xs2 - qr[2];

  // second output: grouped_xyz
  float* out2 = out + (size_t)BATCH * 67 * MQ * KNN;
  out2[(((size_t)b * 3 + 0) * MQ + m) * KNN + kk] = xs0;
  out2[(((size_t)b * 3 + 1) * MQ + m) * KNN + kk] = xs1;
  out2[(((size_t)b * 3 + 2) * MQ + m) * KNN + kk] = xs2;

  // channels 3..66 = gathered features
  const float* fb = feat + (size_t)b * 64 * NPTS + p;
  float* ob = out + (((size_t)b * 67 + 3) * MQ + m) * KNN + kk;
#pragma unroll 4
  for (int c = 0; c < 64; ++c) {
    ob[(size_t)c * MQ * KNN] = fb[(size_t)c * NPTS];
  }
}

// ============================================================================
extern "C" void kernel_launch(void* const* d_in, const int* in_sizes, int n_in,
                              void* d_out, int out_size, void* d_ws, size_t ws_size,
                              hipStream_t stream) {
  (void)in_sizes; (void)n_in; (void)out_size; (void)ws_size;
  const float* xyz     = (const float*)d_in[0];  // (4,16384,3)
  const float* new_xyz = (const float*)d_in[1];  // (4,4096,3)
  const float* feat    = (const float*)d_in[2];  // (4,64,16384)
  float* out = (float*)d_out;
  int* idx_ws = (int*)d_ws;                      // B*M*K ints = 2 MB

  dim3 g1(BATCH * (MQ / WGQ));                   // 256 blocks x 64 threads
  knn_kernel<<<g1, WGQ, 0, stream>>>(xyz, new_xyz, idx_ws);

  dim3 g2((MQ * KNN) / 256, BATCH);              // 512 x 4 blocks x 256 threads
  gather_kernel<<<g2, 256, 0, stream>>>(xyz, new_xyz, feat, idx_ws, out);
}